// RBF_85461259256473
// MI455X (gfx1250) — compile-verified
//
#include <hip/hip_runtime.h>
#include <math.h>

typedef __attribute__((ext_vector_type(2))) float v2f;
typedef __attribute__((ext_vector_type(8))) float v8f;

#define N_ROWS 4096
#define M_ROWS 4096
#define DIM 8
#define KPAD 12   // 8 data + norm + one + 2 pad -> three K=4 WMMA steps

__device__ __forceinline__ float softplus_f(float z) {
    // numerically stable: max(z,0) + log1p(exp(-|z|))
    return fmaxf(z, 0.0f) + log1pf(__expf(-fabsf(z)));
}

// Build augmented operand rows:
//   Ap[i][0..7] =  x[i]/scale, Ap[i][8]=||xs||^2, Ap[i][9]=1,  pad 0
//   Bp[j][0..7] = -2*xx[j]/scale, Bp[j][8]=1, Bp[j][9]=||xxs||^2, pad 0
// so that Ap_row . Bp_row (as K=12 inner product) == squared distance d2.
__global__ __launch_bounds__(256) void rbf_prep_kernel(
    const float* __restrict__ x, const float* __restrict__ xx,
    const float* __restrict__ scale_raw,
    float* __restrict__ Ap, float* __restrict__ Bp)
{
    int i = blockIdx.x * blockDim.x + threadIdx.x;
    if (i >= N_ROWS + M_ROWS) return;

    float rs[DIM];
#pragma unroll
    for (int d = 0; d < DIM; ++d)
        rs[d] = 1.0f / softplus_f(scale_raw[d]);

    if (i < N_ROWS) {
        const float* row = x + (size_t)i * DIM;
        float v[DIM], nrm = 0.0f;
#pragma unroll
        for (int d = 0; d < DIM; ++d) { v[d] = row[d] * rs[d]; nrm += v[d] * v[d]; }
        float* o = Ap + (size_t)i * KPAD;
#pragma unroll
        for (int d = 0; d < DIM; ++d) o[d] = v[d];
        o[8] = nrm; o[9] = 1.0f; o[10] = 0.0f; o[11] = 0.0f;
    } else {
        int j = i - N_ROWS;
        const float* row = xx + (size_t)j * DIM;
        float v[DIM], nrm = 0.0f;
#pragma unroll
        for (int d = 0; d < DIM; ++d) { v[d] = row[d] * rs[d]; nrm += v[d] * v[d]; }
        float* o = Bp + (size_t)j * KPAD;
#pragma unroll
        for (int d = 0; d < DIM; ++d) o[d] = -2.0f * v[d];
        o[8] = 1.0f; o[9] = nrm; o[10] = 0.0f; o[11] = 0.0f;
    }
}

// One wave per 16x16 output tile. 3x v_wmma_f32_16x16x4_f32 accumulate d2,
// epilogue = exp(-0.5*max(d2,0)) * softplus(variance_raw).
__global__ __launch_bounds__(256) void rbf_wmma_kernel(
    const float* __restrict__ Ap, const float* __restrict__ Bp,
    const float* __restrict__ variance_raw, float* __restrict__ out)
{
    const int wavesPerBlock = blockDim.x >> 5;
    const int tile = blockIdx.x * wavesPerBlock + (threadIdx.x >> 5);
    const int tm = tile >> 8;          // 256 tiles along M (rows of out)
    const int tn = tile & 255;         // 256 tiles along N (cols of out)

    const int lane = threadIdx.x & 31;
    const int l = lane & 15;           // position within lane half
    const int h = lane >> 4;           // lane half: 0 -> K {0,1}, 1 -> K {2,3}

    const float* arow = Ap + (size_t)(tm * 16 + l) * KPAD + 2 * h;
    const float* brow = Bp + (size_t)(tn * 16 + l) * KPAD + 2 * h;

    // K-step k0: A vgpr v holds A[l][k0 + 2h + v]  (ISA 16x4 f32 A layout);
    // B is laid out symmetrically since B[k][n] = Bp[n][k].
    v2f a0 = *(const v2f*)(arow + 0);
    v2f b0 = *(const v2f*)(brow + 0);
    v2f a1 = *(const v2f*)(arow + 4);
    v2f b1 = *(const v2f*)(brow + 4);
    v2f a2 = *(const v2f*)(arow + 8);
    v2f b2 = *(const v2f*)(brow + 8);

    v8f c = {};
    c = __builtin_amdgcn_wmma_f32_16x16x4_f32(false, a0, false, b0, (short)0, c, false, false);
    c = __builtin_amdgcn_wmma_f32_16x16x4_f32(false, a1, false, b1, (short)0, c, false, false);
    c = __builtin_amdgcn_wmma_f32_16x16x4_f32(false, a2, false, b2, (short)0, c, false, false);

    const float variance = softplus_f(variance_raw[0]);

    // D layout: vgpr r, lane -> row tm*16 + r + 8*h, col tn*16 + l
    float* ob = out + (size_t)(tm * 16 + 8 * h) * M_ROWS + (tn * 16 + l);
#pragma unroll
    for (int r = 0; r < 8; ++r) {
        float d2 = fmaxf(c[r], 0.0f);
        ob[(size_t)r * M_ROWS] = __expf(-0.5f * d2) * variance;
    }
}

extern "C" void kernel_launch(void* const* d_in, const int* in_sizes, int n_in,
                              void* d_out, int out_size, void* d_ws, size_t ws_size,
                              hipStream_t stream) {
    const float* x            = (const float*)d_in[0];   // (4096, 8)
    const float* xx           = (const float*)d_in[1];   // (4096, 8)
    const float* scale_raw    = (const float*)d_in[2];   // (8,)
    const float* variance_raw = (const float*)d_in[3];   // scalar
    float* out = (float*)d_out;                          // (4096, 4096)

    float* Ap = (float*)d_ws;                            // 4096*12 f32
    float* Bp = Ap + (size_t)N_ROWS * KPAD;              // 4096*12 f32

    {
        int total = N_ROWS + M_ROWS;
        int threads = 256;
        int blocks = (total + threads - 1) / threads;
        rbf_prep_kernel<<<blocks, threads, 0, stream>>>(x, xx, scale_raw, Ap, Bp);
    }
    {
        // 256x256 tiles of 16x16; 8 waves (tiles) per 256-thread block
        int totalTiles = 256 * 256;
        int threads = 256;
        int blocks = totalTiles / 8;
        rbf_wmma_kernel<<<blocks, threads, 0, stream>>>(Ap, Bp, variance_raw, out);
    }
}